// Filter_85495618994849
// MI455X (gfx1250) — compile-verified
//
#include <hip/hip_runtime.h>

// Biquad IIR low-pass via blocked parallel scan + WMMA block filtering.
//   rows B=256, samples T=131072, chunk L=1024, chunks/row C=128.
// Pass1: zero-state end-state per chunk (WMMA block loop, no stores).
// Scan : per-row affine scan over chunk states with M^L (in-place in ws).
// Pass3: re-run each chunk from true initial state, store y (WMMA block loop).

#define NROWS 256
#define NSAMP 131072
#define CHUNK_LEN 1024
#define NCHUNK 128            // NSAMP / CHUNK_LEN
#define NBLOCKS 64            // CHUNK_LEN / 16

typedef __attribute__((ext_vector_type(2))) float v2f;
typedef __attribute__((ext_vector_type(4))) float v4f;
typedef __attribute__((ext_vector_type(8))) float v8f;

__device__ __forceinline__ void biquad_coeffs(const float* __restrict__ cutoff,
                                              const float* __restrict__ resonance,
                                              float& a1, float& a2,
                                              float& cb0, float& cb1, float& cb2) {
    float omega = 2.0f * 3.14159265358979323846f * cutoff[0] / 44100.0f;
    float c = cosf(omega);
    float s = sinf(omega);
    float alpha = s / (2.0f * resonance[0]);
    float a0inv = 1.0f / (1.0f + alpha);
    a1  = -2.0f * c * a0inv;
    a2  = (1.0f - alpha) * a0inv;
    cb0 = 0.5f * (1.0f - c) * a0inv;
    cb1 = (1.0f - c) * a0inv;
    cb2 = cb0;
}

__device__ __forceinline__ v8f wmma4(v2f a, v2f b, v8f c) {
    // D(16x16) = A(16x4) * B(4x16) + C   -- v_wmma_f32_16x16x4_f32
    return __builtin_amdgcn_wmma_f32_16x16x4_f32(false, a, false, b,
                                                 (short)0, c, false, false);
}

// One wave owns 16 streams (= 16 consecutive chunks of one row) and advances
// them 16 samples per iteration with 5 chained WMMAs (K = 16 data + 2 state + 2 pad).
template <bool STORE>
__global__ __launch_bounds__(256) void biquad_pass(
    const float* __restrict__ x,
    const float* __restrict__ cutoff,
    const float* __restrict__ resonance,
    float* __restrict__ y,
    float* __restrict__ ws)   // [NROWS*NCHUNK][2] chunk states
{
    float a1, a2, cb0, cb1, cb2;
    biquad_coeffs(cutoff, resonance, a1, a2, cb0, cb1, cb2);
    // M = [[-a1, b1], [-a2, 0]], v = [b0, b2]

    const int lane = threadIdx.x & 31;
    const int wave = (blockIdx.x * blockDim.x + threadIdx.x) >> 5;
    const int row  = wave >> 3;            // 8 waves per row
    const int j0   = (wave & 7) << 4;      // first stream (chunk) of this wave
    const int m    = lane & 15;            // stream / matrix-row index
    const bool lo  = lane < 16;

    // Impulse response h[d] = (M^d v)[0]; state rows (py,pz)[t] = row0(M^{t+1}).
    float h[16], py[16], pz[16];
    {
        float w0 = cb0, w1 = cb2;
        h[0] = w0;
        for (int d = 1; d < 16; ++d) {
            float n0 = -a1 * w0 + cb1 * w1;
            float n1 = -a2 * w0;
            w0 = n0; w1 = n1;
            h[d] = w0;
        }
        float r0 = -a1, r1 = cb1;          // row0 of M^1
        for (int t = 0; t < 16; ++t) {
            py[t] = r0; pz[t] = r1;
            float n0 = -a1 * r0 - a2 * r1;
            float n1 = cb1 * r0;
            r0 = n0; r1 = n1;
        }
    }
    // A-matrix VGPRs per ISA layout: lanes 0-15 hold (M=lane, K=k0,k0+1),
    // lanes 16-31 hold (M=lane-16, K=k0+2,k0+3).
    v2f A[5];
    for (int g = 0; g < 5; ++g) {
        int k0 = 4 * g + (lo ? 0 : 2);
        float e0, e1;
        // Ahat[t=m, k]
        int k = k0;
        e0 = (k < 16) ? ((k <= m) ? h[m - k] : 0.0f)
                      : ((k == 16) ? py[m] : ((k == 17) ? pz[m] : 0.0f));
        k = k0 + 1;
        e1 = (k < 16) ? ((k <= m) ? h[m - k] : 0.0f)
                      : ((k == 16) ? py[m] : ((k == 17) ? pz[m] : 0.0f));
        A[g].x = e0; A[g].y = e1;
    }

    const size_t base = (size_t)row * NSAMP + (size_t)(j0 + m) * CHUNK_LEN;
    const float* __restrict__ xs = x + base;
    float* __restrict__ ys = y + base;
    const int toff = lo ? 0 : 2;

    float ycur = 0.0f, zcur = 0.0f;
    if (STORE) {
        int chunk = row * NCHUNK + j0 + m;
        ycur = ws[2 * chunk + 0];
        zcur = ws[2 * chunk + 1];
    }

    for (int blk = 0; blk < NBLOCKS; ++blk) {
        const int tb = blk * 16;
        // B-matrix: lanes 0-15 hold (K=k,k+1, N=lane), lanes 16-31 (K=k+2,k+3).
        v2f bv0 = *(const v2f*)(xs + tb + 0  + toff);
        v2f bv1 = *(const v2f*)(xs + tb + 4  + toff);
        v2f bv2 = *(const v2f*)(xs + tb + 8  + toff);
        v2f bv3 = *(const v2f*)(xs + tb + 12 + toff);
        v2f bv4; bv4.x = lo ? ycur : 0.0f; bv4.y = lo ? zcur : 0.0f;

        v8f acc = {0.f, 0.f, 0.f, 0.f, 0.f, 0.f, 0.f, 0.f};
        acc = wmma4(A[0], bv0, acc);
        acc = wmma4(A[1], bv1, acc);
        acc = wmma4(A[2], bv2, acc);
        acc = wmma4(A[3], bv3, acc);
        acc = wmma4(A[4], bv4, acc);

        if (STORE) {
            // C/D layout: VGPR r -> (t=r, lanes 0-15) and (t=r+8, lanes 16-31)
            float* outp = ys + tb + (lo ? 0 : 8);
            v4f o0 = {acc[0], acc[1], acc[2], acc[3]};
            v4f o1 = {acc[4], acc[5], acc[6], acc[7]};
            *(v4f*)(outp + 0) = o0;
            *(v4f*)(outp + 4) = o1;
        }

        // Next-block state: y15 = D[15,n] (acc[7], lanes>=16),
        // z1_15 = b2*x15 - a2*D[14,n]  (x15 = bv3.y on lanes>=16).
        float ynew = acc[7];
        float znew = cb2 * bv3.y - a2 * acc[6];
        int src = (lane & 15) + 16;
        ycur = __shfl(ynew, src, 32);
        zcur = __shfl(znew, src, 32);
    }

    if (!STORE) {
        if (lo) {
            int chunk = row * NCHUNK + j0 + m;
            ws[2 * chunk + 0] = ycur;
            ws[2 * chunk + 1] = zcur;
        }
    }
}

// Affine scan over chunk end-states per row: e_i = M^L e_{i-1} + z_i.
// Rewrites ws[chunk] from zero-state end-state to true initial state, in place.
__global__ __launch_bounds__(256) void biquad_scan(
    const float* __restrict__ cutoff,
    const float* __restrict__ resonance,
    float* __restrict__ ws)
{
    int row = blockIdx.x * blockDim.x + threadIdx.x;
    if (row >= NROWS) return;

    float a1, a2, cb0, cb1, cb2;
    biquad_coeffs(cutoff, resonance, a1, a2, cb0, cb1, cb2);

    // M^L, L = 1024 = 2^10, by repeated squaring of the 2x2
    float m00 = -a1, m01 = cb1, m10 = -a2, m11 = 0.0f;
    for (int i = 0; i < 10; ++i) {
        float t00 = m00 * m00 + m01 * m10;
        float t01 = m00 * m01 + m01 * m11;
        float t10 = m10 * m00 + m11 * m10;
        float t11 = m10 * m01 + m11 * m11;
        m00 = t00; m01 = t01; m10 = t10; m11 = t11;
    }

    float e0 = 0.0f, e1 = 0.0f;           // state before chunk 0
    float* p = ws + (size_t)row * NCHUNK * 2;
    for (int j = 0; j < NCHUNK; ++j) {
        float z0 = p[2 * j + 0];
        float z1 = p[2 * j + 1];
        p[2 * j + 0] = e0;                // true initial state of chunk j
        p[2 * j + 1] = e1;
        float n0 = m00 * e0 + m01 * e1 + z0;
        float n1 = m10 * e0 + m11 * e1 + z1;
        e0 = n0; e1 = n1;
    }
}

extern "C" void kernel_launch(void* const* d_in, const int* in_sizes, int n_in,
                              void* d_out, int out_size, void* d_ws, size_t ws_size,
                              hipStream_t stream) {
    const float* audio     = (const float*)d_in[0];
    const float* cutoff    = (const float*)d_in[1];
    const float* resonance = (const float*)d_in[2];
    float* out = (float*)d_out;
    float* ws  = (float*)d_ws;   // needs NROWS*NCHUNK*2*4 = 256 KB

    // 2048 waves = 256 blocks x 8 waves; each wave owns 16 chunk-streams.
    biquad_pass<false><<<256, 256, 0, stream>>>(audio, cutoff, resonance, out, ws);
    biquad_scan<<<1, 256, 0, stream>>>(cutoff, resonance, ws);
    biquad_pass<true><<<256, 256, 0, stream>>>(audio, cutoff, resonance, out, ws);
}